// ConcatenationAttentionHeads_31860067402326
// MI455X (gfx1250) — compile-verified
//
#include <hip/hip_runtime.h>

#define N_NODES 50000
#define N_EDGES 800000
#define IN_FEAT 128
#define OUT_FEAT 128
#define HEADS 4
#define HEAD_DIM 32
#define NEG_SLOPE 0.2f

typedef __attribute__((ext_vector_type(2))) float v2f;
typedef __attribute__((ext_vector_type(8))) float v8f;

// ---------------------------------------------------------------------------
// Stage 1: h = feat @ W  via V_WMMA_F32_16X16X4_F32 (fp32, full precision).
// One 16x16 output tile per wave32. 3125 blocks x 8 waves = 25000 tiles
// covering [50000 x 128] exactly (no divergence -> EXEC all-1s for WMMA).
// ---------------------------------------------------------------------------
__global__ void __launch_bounds__(256)
gat_gemm_wmma(const float* __restrict__ feat,
              const float* __restrict__ W,
              float* __restrict__ hbuf) {
  const int wave = threadIdx.x >> 5;
  const int lane = threadIdx.x & 31;
  const int tile = blockIdx.x * 8 + wave;   // 0..24999
  const int tm   = tile >> 3;               // row tile 0..3124
  const int tn   = tile & 7;                // col tile 0..7
  const int mr   = lane & 15;               // M (for A) / N (for B)
  const int kq   = (lane >> 4) << 1;        // K sub-offset: 0 or 2

  const float* arow = feat + (tm * 16 + mr) * IN_FEAT;  // A: row-major rows
  const float* bcol = W + tn * 16 + mr;                 // B: column within W

  v8f acc = {};
#pragma unroll
  for (int k = 0; k < IN_FEAT; k += 4) {
    // A 16x4 fp32 layout: lane<16 -> K=k+0,k+1 ; lane>=16 -> K=k+2,k+3
    v2f a = *(const v2f*)(arow + k + kq);
    v2f b;
    b.x = bcol[(k + kq + 0) * OUT_FEAT];
    b.y = bcol[(k + kq + 1) * OUT_FEAT];
    acc = __builtin_amdgcn_wmma_f32_16x16x4_f32(
        /*neg_a=*/false, a, /*neg_b=*/false, b,
        /*c_mod=*/(short)0, acc, /*reuse_a=*/false, /*reuse_b=*/false);
  }

  // D layout: VGPR r holds M = r (lanes 0-15) / r+8 (lanes 16-31), N = lane&15
  const int mbase = tm * 16 + (lane >> 4) * 8;
  float* outp = hbuf + tn * 16 + (lane & 15);
#pragma unroll
  for (int r = 0; r < 8; ++r)
    outp[(mbase + r) * OUT_FEAT] = acc[r];
}

// ---------------------------------------------------------------------------
// Init: out = bias (broadcast), m = -inf, s = 0.
// ---------------------------------------------------------------------------
__global__ void gat_init(float* __restrict__ out, const float* __restrict__ bias,
                         float* __restrict__ m, float* __restrict__ s) {
  int i = blockIdx.x * blockDim.x + threadIdx.x;
  if (i < N_NODES * OUT_FEAT) out[i] = bias[i & (OUT_FEAT - 1)];
  if (i < N_NODES * HEADS) { m[i] = -__builtin_huge_valf(); s[i] = 0.0f; }
}

// ---------------------------------------------------------------------------
// Stage 2: el[n,h] = <h[n,h,:], attn_l[h,:]> ; er likewise.
// ---------------------------------------------------------------------------
__global__ void gat_dots(const float* __restrict__ hbuf,
                         const float* __restrict__ attn_l,
                         const float* __restrict__ attn_r,
                         float* __restrict__ el, float* __restrict__ er) {
  int i = blockIdx.x * blockDim.x + threadIdx.x;  // n*HEADS + h
  if (i >= N_NODES * HEADS) return;
  int n = i >> 2, hh = i & 3;
  const float* hv = hbuf + n * OUT_FEAT + hh * HEAD_DIM;
  const float* al = attn_l + hh * HEAD_DIM;
  const float* ar = attn_r + hh * HEAD_DIM;
  float sl = 0.f, sr = 0.f;
#pragma unroll
  for (int d = 0; d < HEAD_DIM; ++d) { float x = hv[d]; sl += x * al[d]; sr += x * ar[d]; }
  el[i] = sl; er[i] = sr;
}

// Order-independent (deterministic) float atomic max via sign-bit trick.
__device__ __forceinline__ void atomicMaxFloat(float* addr, float v) {
  if (v >= 0.0f) atomicMax((int*)addr, __float_as_int(v));
  else           atomicMin((unsigned int*)addr, __float_as_uint(v));
}

// ---------------------------------------------------------------------------
// Stage 3: e = leaky_relu(el[src] + er[dst]); segment max over dst.
// leaky_relu(v) == max(v, slope*v) for slope < 1.
// ---------------------------------------------------------------------------
__global__ void gat_edge_score(const int* __restrict__ src, const int* __restrict__ dst,
                               const float* __restrict__ el, const float* __restrict__ er,
                               float* __restrict__ ebuf, float* __restrict__ mbuf) {
  int i = blockIdx.x * blockDim.x + threadIdx.x;  // eid*HEADS + h
  if (i >= N_EDGES * HEADS) return;
  int eid = i >> 2, hh = i & 3;
  float v = el[src[eid] * HEADS + hh] + er[dst[eid] * HEADS + hh];
  v = fmaxf(v, NEG_SLOPE * v);
  ebuf[i] = v;
  atomicMaxFloat(mbuf + dst[eid] * HEADS + hh, v);
}

// ---------------------------------------------------------------------------
// Stage 4: ex = exp(e - m[dst]); segment sum over dst.
// ---------------------------------------------------------------------------
__global__ void gat_edge_exp(const int* __restrict__ dst,
                             const float* __restrict__ mbuf,
                             float* __restrict__ ebuf, float* __restrict__ sbuf) {
  int i = blockIdx.x * blockDim.x + threadIdx.x;
  if (i >= N_EDGES * HEADS) return;
  int eid = i >> 2, hh = i & 3;
  int j = dst[eid] * HEADS + hh;
  float ex = expf(ebuf[i] - mbuf[j]);
  ebuf[i] = ex;
  atomicAdd(sbuf + j, ex);
}

// ---------------------------------------------------------------------------
// Stage 5: out[dst,h,:] += (ex / s[dst,h]) * h[src,h,:]  (scatter-add)
// h and out both live in 192MB L2, so gathers+atomics resolve on-chip.
// ---------------------------------------------------------------------------
__global__ void gat_edge_aggregate(const float* __restrict__ ebuf,
                                   const float* __restrict__ sbuf,
                                   const int* __restrict__ src, const int* __restrict__ dst,
                                   const float* __restrict__ hbuf, float* __restrict__ out) {
  int i = blockIdx.x * blockDim.x + threadIdx.x;  // eid*HEADS + h
  if (i >= N_EDGES * HEADS) return;
  int eid = i >> 2, hh = i & 3;
  int dn = dst[eid], sn = src[eid];
  float alpha = ebuf[i] / sbuf[dn * HEADS + hh];
  const float4* hv = (const float4*)(hbuf + sn * OUT_FEAT + hh * HEAD_DIM);
  float* ov = out + dn * OUT_FEAT + hh * HEAD_DIM;
#pragma unroll
  for (int q = 0; q < HEAD_DIM / 4; ++q) {
    float4 v = hv[q];
    atomicAdd(ov + q * 4 + 0, alpha * v.x);
    atomicAdd(ov + q * 4 + 1, alpha * v.y);
    atomicAdd(ov + q * 4 + 2, alpha * v.z);
    atomicAdd(ov + q * 4 + 3, alpha * v.w);
  }
}

extern "C" void kernel_launch(void* const* d_in, const int* in_sizes, int n_in,
                              void* d_out, int out_size, void* d_ws, size_t ws_size,
                              hipStream_t stream) {
  (void)in_sizes; (void)n_in; (void)out_size; (void)ws_size;
  const float* feat   = (const float*)d_in[0];
  const int*   src    = (const int*)  d_in[1];
  const int*   dst    = (const int*)  d_in[2];
  const float* W      = (const float*)d_in[3];
  const float* attn_l = (const float*)d_in[4];
  const float* attn_r = (const float*)d_in[5];
  const float* bias   = (const float*)d_in[6];
  float* out = (float*)d_out;

  // Workspace layout (10.4M floats = 41.6 MB)
  float* hbuf = (float*)d_ws;                           // N*128
  float* ebuf = hbuf + (size_t)N_NODES * OUT_FEAT;      // E*H
  float* mbuf = ebuf + (size_t)N_EDGES * HEADS;         // N*H
  float* sbuf = mbuf + (size_t)N_NODES * HEADS;         // N*H
  float* elb  = sbuf + (size_t)N_NODES * HEADS;         // N*H
  float* erb  = elb  + (size_t)N_NODES * HEADS;         // N*H

  const int TPB = 256;
  gat_init<<<(N_NODES * OUT_FEAT + TPB - 1) / TPB, TPB, 0, stream>>>(out, bias, mbuf, sbuf);
  gat_gemm_wmma<<<3125, 256, 0, stream>>>(feat, W, hbuf);
  gat_dots<<<(N_NODES * HEADS + TPB - 1) / TPB, TPB, 0, stream>>>(hbuf, attn_l, attn_r, elb, erb);
  gat_edge_score<<<(N_EDGES * HEADS + TPB - 1) / TPB, TPB, 0, stream>>>(src, dst, elb, erb, ebuf, mbuf);
  gat_edge_exp<<<(N_EDGES * HEADS + TPB - 1) / TPB, TPB, 0, stream>>>(dst, mbuf, ebuf, sbuf);
  gat_edge_aggregate<<<(N_EDGES * HEADS + TPB - 1) / TPB, TPB, 0, stream>>>(ebuf, sbuf, src, dst, hbuf, out);
}